// TnoBlockFFT_67207648248371
// MI455X (gfx1250) — compile-verified
//
#include <hip/hip_runtime.h>

// ----------------------------------------------------------------------------
// TNO block-FFT convolution for MI455X (gfx1250), fp32 via V_WMMA_F32_16X16X4_F32
//
//   o[b,:,c] = first 4096 samples of IFFT_8192( FFT_8192(pad(x[b,:,c])) *
//                                               FFT_8192(t[:,c]) )
//
// Four-step FFT, N = 8192 = R(128) x C(64):
//   n = c + 64*r ; k = k_r + 128*k_c
//   stage1: Y[k_r,c]  = sum_r W128[k_r,r] * x[c + 64 r]       (GEMM, K=64 for x)
//   twiddle: Y *= w_N^{-k_r c}
//   stage2: X[k_r,k_c]= sum_c Y[k_r,c] * W64[c,k_c]           (GEMM)
// Inverse mirrored (conjugate twiddles, 1/N, real part, n2 < 64 only).
//
// Sequence staging (strided channel gather) is done by the Tensor Data Mover
// (TENSOR_LOAD_TO_LDS, 2D tile 1 x rows, stride 512 elems) when available.
//
// d_ws requirement: 512*8192*2 floats = 33,554,432 bytes (FFT(t) planes).
// ----------------------------------------------------------------------------

#define TWO_PI 6.28318530717958647692f

typedef float v2f __attribute__((ext_vector_type(2)));
typedef float v8f __attribute__((ext_vector_type(8)));

#define WMMA_F32(a, b, c)                                                     \
  __builtin_amdgcn_wmma_f32_16x16x4_f32(false, (a), false, (b), (short)0,     \
                                        (c), false, false)

// Single LDS arena (one shared symbol -> LDS offset 0, so TDM descriptors can
// carry absolute LDS byte addresses).  Layout (floats):
//   [0      , 8192 )  Are
//   [8192   , 16384)  Aim
//   [16384  , 24576)  Bre   (aliased by x0 / t0 staging buffer)
//   [24576  , 32768)  Bim
//   [32768  , 32896)  tabc
//   [32896  , 33024)  tabs
#define SMEM_FLOATS (4 * 8192 + 256)
#define STAGE_LDS_BYTE_OFF (16384u * 4u)

// ---------------------------------------------------------------------------
// TDM staging: load rows 4-byte elements t[row*512] into LDS (contiguous).
// D# bitfields per CDNA5 ISA:
//   g0: count=1 | lds_addr | global_addr[56:0] | type=2
//   g1: data_size=4B, tensor_dim0=512, tensor_dim1=rows, tile=(1, rows),
//       tensor_dim0_stride=512
// Toolchain probe: this clang exposes the 6-arg builtin
//   (uint32x4 g0, int32x8 g1, int32x4 g2, int32x4 g3, int32x8, i32 cpol)
// ---------------------------------------------------------------------------
#if __has_builtin(__builtin_amdgcn_tensor_load_to_lds)
#define USE_TDM 1
typedef unsigned int v4u __attribute__((ext_vector_type(4)));
typedef int v8i __attribute__((ext_vector_type(8)));
typedef int v4i __attribute__((ext_vector_type(4)));

__device__ __forceinline__ void tdm_load_column(const float* gbase,
                                                unsigned lds_byte_off,
                                                int rows) {
  const unsigned long long ga = (unsigned long long)(uintptr_t)gbase;
  v4u g0;
  g0[0] = 1u;                                   // count=1, user descriptor
  g0[1] = lds_byte_off;                         // lds_addr
  g0[2] = (unsigned)(ga & 0xffffffffu);         // global_addr[31:0]
  g0[3] = (unsigned)((ga >> 32) & 0x01ffffffu)  // global_addr[56:32]
          | (2u << 30);                         // type = 2 ("image")
  v8i g1;
  g1[0] = 0x00020000;                     // data_size = 2 (4 bytes)
  g1[1] = (int)(512u << 16);              // tensor_dim0 = 512 (low 16)
  g1[2] = (int)((unsigned)rows << 16);    // dim0 hi | tensor_dim1 low16 = rows
  g1[3] = (int)(1u << 16);                // dim1 hi | tile_dim0 = 1
  g1[4] = rows;                           // tile_dim1 = rows, tile_dim2 = 0
  g1[5] = 512;                            // tensor_dim0_stride = 512
  g1[6] = 0;
  g1[7] = 0;
  v4i z4 = {0, 0, 0, 0};
  v8i z8 = {0, 0, 0, 0, 0, 0, 0, 0};
  __builtin_amdgcn_tensor_load_to_lds(g0, g1, z4, z4, z8, 0);
}

__device__ __forceinline__ void tdm_wait0() {
#if __has_builtin(__builtin_amdgcn_s_wait_tensorcnt)
  __builtin_amdgcn_s_wait_tensorcnt(0);
#else
  asm volatile("s_wait_tensorcnt 0x0" ::: "memory");
#endif
}
#else
#define USE_TDM 0
#endif

// ---------------------------------------------------------------------------
// Stage kind A: Y(128x64, complex) = W128[:, :K] * X, X[r,c] = x0[c + 64 r],
// x0 real.  W128[kr,r] = cos(2pi kr r/128) - i sin(...).
// Each wave owns one row tile (tm = wave) and all 4 column tiles, so the
// W-matrix fragment (table gather) is reused across 4 column tiles (8 wmma).
// ---------------------------------------------------------------------------
__device__ __forceinline__ void stage_dft_left_real(
    const float* x0, float* Yre, float* Yim,
    const float* tabc, const float* tabs, int K, int wave, int lane) {
  const int m16 = lane & 15;
  const int hi  = lane >> 4;
  const int ksteps = K >> 2;
  const int tm = wave;               // 8 waves <-> 8 row tiles (k_r)
  const int m  = tm * 16 + m16;
  v8f accC[4] = {{}, {}, {}, {}};
  v8f accS[4] = {{}, {}, {}, {}};
  for (int ks = 0; ks < ksteps; ++ks) {
    const int ka = ks * 4 + 2 * hi;  // this lane's K pair (ISA A layout)
    v2f ac, as;
    ac[0] = tabc[(m * ka) & 127];
    ac[1] = tabc[(m * (ka + 1)) & 127];
    as[0] = tabs[(m * ka) & 127];
    as[1] = tabs[(m * (ka + 1)) & 127];
#pragma unroll
    for (int tn = 0; tn < 4; ++tn) {
      const int n = tn * 16 + m16;
      v2f b;
      b[0] = x0[n + 64 * ka];
      b[1] = x0[n + 64 * (ka + 1)];
      accC[tn] = WMMA_F32(ac, b, accC[tn]);
      accS[tn] = WMMA_F32(as, b, accS[tn]);
    }
  }
#pragma unroll
  for (int tn = 0; tn < 4; ++tn) {
    const int n = tn * 16 + m16;
#pragma unroll
    for (int j = 0; j < 8; ++j) {    // C/D layout: VGPR j -> row j (+8 for hi)
      const int row = tm * 16 + j + 8 * hi;
      Yre[row * 64 + n] = accC[tn][j];
      Yim[row * 64 + n] = -accS[tn][j];
    }
  }
}

// ---------------------------------------------------------------------------
// Stage kind B: O(128x64) = A(128x64, complex) * W64,
// W64[k,n] = cos(2pi k n/64) + i*ssign*sin(2pi k n/64)
//   ssign = -1 : forward DFT, ssign = +1 : inverse (conjugate) DFT
// Wave owns one row tile; data fragments (b64 LDS loads) reused across 4
// column tiles (16 wmma per k-step from just 2 b64 loads + 16 table gathers).
// ---------------------------------------------------------------------------
__device__ __forceinline__ void stage_dft_right(
    const float* Are, const float* Aim, float* Ore, float* Oim,
    const float* tabc, const float* tabs, float ssign, int wave, int lane) {
  const int m16 = lane & 15;
  const int hi  = lane >> 4;
  const int tm  = wave;
  const int m   = tm * 16 + m16;
  v8f accRe[4] = {{}, {}, {}, {}};
  v8f accIm[4] = {{}, {}, {}, {}};
  for (int ks = 0; ks < 16; ++ks) {
    const int ka = ks * 4 + 2 * hi;
    v2f ar = *reinterpret_cast<const v2f*>(&Are[m * 64 + ka]);
    v2f ai = *reinterpret_cast<const v2f*>(&Aim[m * 64 + ka]);
#pragma unroll
    for (int tn = 0; tn < 4; ++tn) {
      const int n = tn * 16 + m16;
      v2f bc, bs, bsn;
      bc[0] = tabc[(2 * ka * n) & 127];
      bc[1] = tabc[(2 * (ka + 1) * n) & 127];
      const float s0 = tabs[(2 * ka * n) & 127];
      const float s1 = tabs[(2 * (ka + 1) * n) & 127];
      bs[0] = ssign * s0;  bs[1] = ssign * s1;
      bsn[0] = -bs[0];     bsn[1] = -bs[1];
      // O_re = A_re*C - A_im*(sS) ; O_im = A_im*C + A_re*(sS)
      accRe[tn] = WMMA_F32(ar, bc, accRe[tn]);
      accRe[tn] = WMMA_F32(ai, bsn, accRe[tn]);
      accIm[tn] = WMMA_F32(ai, bc, accIm[tn]);
      accIm[tn] = WMMA_F32(ar, bs, accIm[tn]);
    }
  }
#pragma unroll
  for (int tn = 0; tn < 4; ++tn) {
    const int n = tn * 16 + m16;
#pragma unroll
    for (int j = 0; j < 8; ++j) {
      const int row = tm * 16 + j + 8 * hi;
      Ore[row * 64 + n] = accRe[tn][j];
      Oim[row * 64 + n] = accIm[tn][j];
    }
  }
}

// ---------------------------------------------------------------------------
// Final inverse stage (real output, truncated):
// o[n1 + 64*n2] = (1/8192) * sum_kr [ cos(2pi n2 kr/128)*Yre[kr,n1]
//                                   - sin(2pi n2 kr/128)*Yim[kr,n1] ], n2<64
// 16 tiles (4x4): wave owns tm = wave&3 and two column tiles; A fragment
// (table gather) shared across both.
// ---------------------------------------------------------------------------
__device__ __forceinline__ void stage_ifft_final(
    const float* Yre, const float* Yim, float* out, long obase,
    const float* tabc, const float* tabs, int wave, int lane) {
  const int m16 = lane & 15;
  const int hi  = lane >> 4;
  const int tm  = wave & 3;
  const int tn0 = (wave >> 2) * 2;
  const int m   = tm * 16 + m16;     // n2 row
  v8f acc[2] = {{}, {}};
  for (int ks = 0; ks < 32; ++ks) {  // K = 128 (k_r)
    const int ka = ks * 4 + 2 * hi;
    v2f ac, asn;
    ac[0]  = tabc[(m * ka) & 127];
    ac[1]  = tabc[(m * (ka + 1)) & 127];
    asn[0] = -tabs[(m * ka) & 127];
    asn[1] = -tabs[(m * (ka + 1)) & 127];
#pragma unroll
    for (int tt = 0; tt < 2; ++tt) {
      const int n = (tn0 + tt) * 16 + m16;   // n1 col
      v2f bre, bim;
      bre[0] = Yre[ka * 64 + n];
      bre[1] = Yre[(ka + 1) * 64 + n];
      bim[0] = Yim[ka * 64 + n];
      bim[1] = Yim[(ka + 1) * 64 + n];
      acc[tt] = WMMA_F32(ac, bre, acc[tt]);
      acc[tt] = WMMA_F32(asn, bim, acc[tt]);
    }
  }
#pragma unroll
  for (int tt = 0; tt < 2; ++tt) {
    const int n = (tn0 + tt) * 16 + m16;
#pragma unroll
    for (int j = 0; j < 8; ++j) {
      const int n2 = tm * 16 + j + 8 * hi;
      const int nn = n + 64 * n2;            // output sample index
      out[obase + (long)nn * 512] = acc[tt][j] * (1.0f / 8192.0f);
    }
  }
}

// Pointwise twiddle: Y[kr,c] *= cos(2pi kr c/8192) + i*sgn*sin(...)
__device__ __forceinline__ void twiddle_8192(float* Re, float* Im, float sgn,
                                             int tid) {
  for (int e = tid; e < 8192; e += 256) {
    const int kr = e >> 6, c = e & 63;
    float s, cth;
    __sincosf((float)(kr * c) * (TWO_PI / 8192.0f), &s, &cth);
    s *= sgn;
    const float re = Re[e], im = Im[e];
    Re[e] = re * cth - im * s;
    Im[e] = im * cth + re * s;
  }
}

__device__ __forceinline__ void init_tables(float* tabc, float* tabs, int tid) {
  if (tid < 128) {
    float s, c;
    __sincosf((float)tid * (TWO_PI / 128.0f), &s, &c);
    tabc[tid] = c;
    tabs[tid] = s;
  }
}

// ---------------------------------------------------------------------------
// Kernel 1: per-channel FFT(t) -> workspace planes (channel-major, k-linear)
// ---------------------------------------------------------------------------
__global__ __launch_bounds__(256) void tno_tfft_kernel(
    const float* __restrict__ t, float* __restrict__ wsTre,
    float* __restrict__ wsTim) {
  __shared__ __align__(16) float smem[SMEM_FLOATS];  // ~129 KB
  float* Are  = smem;
  float* Aim  = smem + 8192;
  float* Bre  = smem + 16384;
  float* Bim  = smem + 24576;
  float* t0   = Bre;  // staging aliases B planes (dead until stage 2)
  float* tabc = smem + 32768;
  float* tabs = smem + 32896;

  const int ch   = blockIdx.x;
  const int tid  = threadIdx.x;
  const int wave = tid >> 5;
  const int lane = tid & 31;

#if USE_TDM
  if (wave == 0) tdm_load_column(t + ch, STAGE_LDS_BYTE_OFF, 8192);
#endif
  init_tables(tabc, tabs, tid);
#if USE_TDM
  if (wave == 0) tdm_wait0();
#else
  for (int j = tid; j < 8192; j += 256) t0[j] = t[(long)j * 512 + ch];
#endif
  __syncthreads();

  stage_dft_left_real(t0, Are, Aim, tabc, tabs, /*K=*/128, wave, lane);
  __syncthreads();
  twiddle_8192(Are, Aim, -1.0f, tid);
  __syncthreads();
  stage_dft_right(Are, Aim, Bre, Bim, tabc, tabs, -1.0f, wave, lane);
  __syncthreads();

  // store k-linear (k = kr + 128*kc): coalesced for kernel 2
  for (int k = tid; k < 8192; k += 256) {
    const int kr = k & 127, kc = k >> 7;
    wsTre[(long)ch * 8192 + k] = Bre[kr * 64 + kc];
    wsTim[(long)ch * 8192 + k] = Bim[kr * 64 + kc];
  }
}

// ---------------------------------------------------------------------------
// Kernel 2: full FFT-conv pipeline, one workgroup per (batch, channel)
// ---------------------------------------------------------------------------
__global__ __launch_bounds__(256) void tno_conv_kernel(
    const float* __restrict__ x, const float* __restrict__ wsTre,
    const float* __restrict__ wsTim, float* __restrict__ out) {
  __shared__ __align__(16) float smem[SMEM_FLOATS];  // ~129 KB
  float* Are  = smem;
  float* Aim  = smem + 8192;
  float* Bre  = smem + 16384;
  float* Bim  = smem + 24576;
  float* x0   = Bre;  // x0 (4096 floats) aliases the dead B plane
  float* tabc = smem + 32768;
  float* tabs = smem + 32896;

  const int blk  = blockIdx.x;
  const int b    = blk >> 9;
  const int ch   = blk & 511;
  const int tid  = threadIdx.x;
  const int wave = tid >> 5;
  const int lane = tid & 31;

  const float* Tre = wsTre + (long)ch * 8192;
  const float* Tim = wsTim + (long)ch * 8192;

#if USE_TDM
  if (wave == 0)
    tdm_load_column(x + (long)b * 4096 * 512 + ch, STAGE_LDS_BYTE_OFF, 4096);
#endif
  init_tables(tabc, tabs, tid);
  // warm the spectral planes into cache while the forward FFT runs
  __builtin_prefetch(Tre + tid * 32, 0, 1);
  __builtin_prefetch(Tim + tid * 32, 0, 1);
#if USE_TDM
  if (wave == 0) tdm_wait0();
#else
  {
    const float* xseq = x + (long)b * 4096 * 512 + ch;
    for (int j = tid; j < 4096; j += 256) x0[j] = xseq[(long)j * 512];
  }
#endif
  __syncthreads();

  // forward FFT (zero-pad exploited: stage-1 reduction depth 64)
  stage_dft_left_real(x0, Are, Aim, tabc, tabs, /*K=*/64, wave, lane);
  __syncthreads();
  twiddle_8192(Are, Aim, -1.0f, tid);
  __syncthreads();
  stage_dft_right(Are, Aim, Bre, Bim, tabc, tabs, -1.0f, wave, lane);
  __syncthreads();

  // spectral multiply: X[k] *= T[k], k = kr + 128*kc (coalesced global reads)
  for (int k = tid; k < 8192; k += 256) {
    const int kr = k & 127, kc = k >> 7;
    const int e  = kr * 64 + kc;
    const float xr = Bre[e], xi = Bim[e];
    const float tr = Tre[k], ti = Tim[k];
    Bre[e] = xr * tr - xi * ti;
    Bim[e] = xr * ti + xi * tr;
  }
  __syncthreads();

  // inverse FFT (conjugate stages), truncated real output
  stage_dft_right(Bre, Bim, Are, Aim, tabc, tabs, +1.0f, wave, lane);
  __syncthreads();
  twiddle_8192(Are, Aim, +1.0f, tid);
  __syncthreads();
  stage_ifft_final(Are, Aim, out, (long)b * 4096 * 512 + ch, tabc, tabs, wave,
                   lane);
}

// ---------------------------------------------------------------------------
extern "C" void kernel_launch(void* const* d_in, const int* in_sizes, int n_in,
                              void* d_out, int out_size, void* d_ws,
                              size_t ws_size, hipStream_t stream) {
  const float* x = (const float*)d_in[0];   // (4, 4096, 512) f32
  const float* t = (const float*)d_in[1];   // (8192, 512) f32
  float* out = (float*)d_out;               // (4, 4096, 512) f32

  // workspace: FFT(t) planes, channel-major.  Requires 33,554,432 bytes.
  float* wsTre = (float*)d_ws;
  float* wsTim = wsTre + (size_t)512 * 8192;

  tno_tfft_kernel<<<512, 256, 0, stream>>>(t, wsTre, wsTim);
  tno_conv_kernel<<<2048, 256, 0, stream>>>(x, wsTre, wsTim, out);
}